// Arcface_37701222924459
// MI455X (gfx1250) — compile-verified
//
#include <hip/hip_runtime.h>
#include <hip/hip_bf16.h>
#include <math.h>

typedef float v2f __attribute__((ext_vector_type(2)));
typedef float v8f __attribute__((ext_vector_type(8)));

#define BATCH 512
#define FDIM  512
#define S_SCALE 64.0f
#define MARGIN  0.5f
#define NORM_EPS 1e-12f

// group-of-32 XOR swizzle: offset = {1'b0, xor[14:10], or[9:5], and[4:0]}
template <int IMM>
__device__ __forceinline__ float swz_xor_add(float x) {
    int y = __builtin_amdgcn_ds_swizzle(__float_as_int(x), IMM);
    return x + __int_as_float(y);
}

// ---------------- K0: zero the atomic accumulators (d, Z) ----------------
__global__ void k0_zero(float* p, int n) {
    int i = blockIdx.x * blockDim.x + threadIdx.x;
    if (i < n) p[i] = 0.0f;
}

// ---------------- K1: row L2-normalize feature, store K-major K-pair-packed
// Apk (as float2 units): Apk[kk*512 + m] = { feat_n[m][2kk], feat_n[m][2kk+1] }
__global__ void k1_normalize_pack(const float* __restrict__ feat,
                                  float* __restrict__ Apk) {
    const int m   = blockIdx.x;      // 512 blocks, one row each
    const int tid = threadIdx.x;     // 256 threads
    __shared__ float row[FDIM];
    __shared__ float red[256];

    float f0 = feat[m * FDIM + tid];
    float f1 = feat[m * FDIM + tid + 256];
    row[tid]       = f0;
    row[tid + 256] = f1;
    red[tid] = f0 * f0 + f1 * f1;
    __syncthreads();
    for (int off = 128; off > 0; off >>= 1) {
        if (tid < off) red[tid] += red[tid + off];
        __syncthreads();
    }
    const float inv = 1.0f / fmaxf(sqrtf(red[0]), NORM_EPS);

    // thread tid handles K-pair kk = tid
    const int kk = tid;
    v2f o;
    o.x = row[2 * kk]     * inv;
    o.y = row[2 * kk + 1] * inv;
    *(v2f*)(Apk + (size_t)kk * (2 * FDIM) + 2 * m) = o;
}

// ---------------- K2: GEMM (512 x C, K=512) via V_WMMA_F32_16X16X4_F32 ----
// block = 256 threads = 8 waves; wave tile = 32M x 64N; block tile = 256M x 64N
// grid = ( ceil(C/64), 2 )
// N-edge handling: clamp column addresses (always-legal loads, no predication
// in the K-loop); mask OOB lanes once in the epilogue instead.
__global__ void __launch_bounds__(256)
k2_gemm(const float* __restrict__ Apk,
        const float* __restrict__ hdr,
        float* __restrict__ logits,
        float* __restrict__ dabs,
        int C) {
    const int lane = threadIdx.x & 31;
    const int wave = threadIdx.x >> 5;
    const int half = lane >> 4;          // 0: lanes 0-15, 1: lanes 16-31
    const int ln   = lane & 15;
    const int m0   = blockIdx.y * 256 + wave * 32;
    const int nb   = blockIdx.x * 64;

    // loop-invariant, clamped per-lane column indices
    bool cv[4];
    int  colc[4];
#pragma unroll
    for (int ns = 0; ns < 4; ++ns) {
        const int col = nb + ns * 16 + ln;
        cv[ns]   = (col < C);
        colc[ns] = cv[ns] ? col : (C - 1);
    }

    v8f acc[2][4];
#pragma unroll
    for (int i = 0; i < 2; ++i)
#pragma unroll
        for (int j = 0; j < 4; ++j) acc[i][j] = {};

    const float* __restrict__ aPtr = Apk + (size_t)half * (2 * FDIM);
    const float* __restrict__ hPtr = hdr + (size_t)(2 * half) * C;

    for (int ki = 0; ki < FDIM / 4; ++ki) {
        // A operands: one coalesced 8B load per 16M subtile
        const float* ab = aPtr + (size_t)(ki * 2) * (2 * FDIM);
        v2f a0 = *(const v2f*)(ab + 2 * (m0 + ln));
        v2f a1 = *(const v2f*)(ab + 2 * (m0 + 16 + ln));

        const float* h0 = hPtr + (size_t)(ki * 4) * C;
#pragma unroll
        for (int ns = 0; ns < 4; ++ns) {
            v2f b;
            b.x = h0[colc[ns]];
            b.y = h0[colc[ns] + C];
            acc[0][ns] = __builtin_amdgcn_wmma_f32_16x16x4_f32(
                false, a0, false, b, (short)0, acc[0][ns], false, false);
            acc[1][ns] = __builtin_amdgcn_wmma_f32_16x16x4_f32(
                false, a1, false, b, (short)0, acc[1][ns], false, false);
        }
    }

    // ---------------- epilogue ----------------
#pragma unroll
    for (int ms = 0; ms < 2; ++ms) {
        const int rbase = m0 + ms * 16 + half * 8;   // C/D: vgpr j -> row j (+8 for hi half)

        // stores: one exec toggle per 16x16 subtile, pointer-incremented rows
#pragma unroll
        for (int ns = 0; ns < 4; ++ns) {
            if (cv[ns]) {
                const int col = nb + ns * 16 + ln;
                size_t idx = (size_t)rbase * C + col;
#pragma unroll
                for (int j = 0; j < 8; ++j) {
                    logits[idx] = acc[ms][ns][j];
                    idx += (size_t)C;
                }
            }
        }

        // per-row sum(|x|): accumulate the 4 N-subtiles per lane first,
        // then one 16-lane XOR-swizzle tree per row
#pragma unroll
        for (int j = 0; j < 8; ++j) {
            float s = 0.0f;
#pragma unroll
            for (int ns = 0; ns < 4; ++ns)
                s += cv[ns] ? fabsf(acc[ms][ns][j]) : 0.0f;
            s = swz_xor_add<0x041F>(s);   // xor 1
            s = swz_xor_add<0x081F>(s);   // xor 2
            s = swz_xor_add<0x101F>(s);   // xor 4
            s = swz_xor_add<0x201F>(s);   // xor 8
            if (ln == 0) atomicAdd(&dabs[rbase + j], s);
        }
    }
}

// ---------------- K3: per-row sum of exp(S*x/d), target class excluded ----
__global__ void k3_expsum(const float* __restrict__ logits,
                          const float* __restrict__ dabs,
                          const long long* __restrict__ label,
                          float* __restrict__ Z, int C) {
    const int row = blockIdx.x;
    const int lab = (int)label[row];
    const float scale = S_SCALE / dabs[row];
    const float* lr = logits + (size_t)row * C;

    float s = 0.0f;
    // C is even in practice -> row base is 8B aligned; use float2 loads
    const int n2 = C >> 1;
    const v2f* lr2 = (const v2f*)lr;
    for (int i = threadIdx.x; i < n2; i += blockDim.x) {
        const v2f x = lr2[i];
        const int c0 = 2 * i;
        const float e0 = expf(scale * x.x);
        const float e1 = expf(scale * x.y);
        s += (c0     == lab) ? 0.0f : e0;
        s += (c0 + 1 == lab) ? 0.0f : e1;
    }
    if ((C & 1) && threadIdx.x == 0) {          // safety tail (unused for C=85742)
        const int c = C - 1;
        s += (c == lab) ? 0.0f : expf(scale * lr[c]);
    }

    __shared__ float red[256];
    red[threadIdx.x] = s;
    __syncthreads();
    for (int off = 128; off > 0; off >>= 1) {
        if (threadIdx.x < off) red[threadIdx.x] += red[threadIdx.x + off];
        __syncthreads();
    }
    if (threadIdx.x == 0) Z[row] = red[0];
}

// ---------------- K4: margin injection + LSE + mean -----------------------
__global__ void k4_loss(const float* __restrict__ logits,
                        const float* __restrict__ dabs,
                        const float* __restrict__ Z,
                        const long long* __restrict__ label,
                        float* __restrict__ out, int C) {
    const int b = threadIdx.x;     // one block, 512 threads
    const int lab = (int)label[b];
    const float d = dabs[b];
    float t = logits[(size_t)b * C + lab] / d;
    t = fminf(1.0f, fmaxf(-1.0f, t));
    const float tm = cosf(acosf(t) + MARGIN);
    const float Zb = Z[b] + expf(S_SCALE * tm);   // S*x <= 64 -> no overflow
    float contrib = logf(Zb) - S_SCALE * tm;

    __shared__ float red[BATCH];
    red[b] = contrib;
    __syncthreads();
    for (int off = 256; off > 0; off >>= 1) {
        if (b < off) red[b] += red[b + off];
        __syncthreads();
    }
    if (b == 0) out[0] = red[0] / (float)BATCH;
}

extern "C" void kernel_launch(void* const* d_in, const int* in_sizes, int n_in,
                              void* d_out, int out_size, void* d_ws, size_t ws_size,
                              hipStream_t stream) {
    const float*      feature = (const float*)d_in[0];
    const float*      header  = (const float*)d_in[1];
    const long long*  label   = (const long long*)d_in[2];
    float*            out     = (float*)d_out;

    const int C = in_sizes[1] / FDIM;     // 85742

    // workspace layout (floats)
    float* ws      = (float*)d_ws;
    float* Apk     = ws;                              // 512*512       (1 MB)
    float* dabs    = ws + (size_t)FDIM * BATCH;       // 512
    float* Z       = dabs + BATCH;                    // 512
    float* logits  = Z + BATCH;                       // 512*C         (~176 MB)

    k0_zero<<<(2 * BATCH + 255) / 256, 256, 0, stream>>>(dabs, 2 * BATCH);
    k1_normalize_pack<<<BATCH, 256, 0, stream>>>(feature, Apk);

    dim3 g2((C + 63) / 64, 2, 1);
    k2_gemm<<<g2, 256, 0, stream>>>(Apk, header, logits, dabs, C);

    k3_expsum<<<BATCH, 256, 0, stream>>>(logits, dabs, label, Z, C);
    k4_loss<<<1, BATCH, 0, stream>>>(logits, dabs, Z, label, out, C);
}